// Generator_79989470921097
// MI455X (gfx1250) — compile-verified
//
#include <hip/hip_runtime.h>

typedef __attribute__((ext_vector_type(2))) float v2f;
typedef __attribute__((ext_vector_type(8))) float v8f;

#define FEAT 128  // inner feature dim (K) for both layers

// ---------------------------------------------------------------------------
// Edge scatter: one wave (32 lanes) per edge. Each lane moves 4 consecutive
// floats (float4 -> GLOBAL_LOAD_B128) of the 128-float source row and
// accumulates with global_atomic_add_f32. Lane 0 counts the edge.
// ---------------------------------------------------------------------------
__global__ void __launch_bounds__(256)
sage_scatter(const float* __restrict__ xsrc,
             const int* __restrict__ src,
             const int* __restrict__ dst,
             float* __restrict__ agg,
             float* __restrict__ cnt,
             int E)
{
    int wave = (int)((blockIdx.x * blockDim.x + threadIdx.x) >> 5);
    int lane = threadIdx.x & 31;
    if (wave >= E) return;
    int s = src[wave];
    int d = dst[wave];
    const float4* row = (const float4*)(xsrc + (size_t)s * FEAT);
    float4 v = row[lane];
    float* o = agg + (size_t)d * FEAT + lane * 4;
    atomicAdd(o + 0, v.x);
    atomicAdd(o + 1, v.y);
    atomicAdd(o + 2, v.z);
    atomicAdd(o + 3, v.w);
    if (lane == 0) atomicAdd(cnt + d, 1.0f);
}

// ---------------------------------------------------------------------------
// mean = agg / max(cnt, 1)   (isolated targets -> 0, matching the reference)
// ---------------------------------------------------------------------------
__global__ void __launch_bounds__(256)
sage_mean_div(float* __restrict__ agg, const float* __restrict__ cnt, int total)
{
    int i = blockIdx.x * blockDim.x + threadIdx.x;
    if (i >= total) return;
    float c = cnt[i >> 7];                 // FEAT == 128
    agg[i] *= 1.0f / fmaxf(c, 1.0f);
}

// ---------------------------------------------------------------------------
// Fused SAGE linear: out[m,n] = relu( sum_k A1[m,k]*W1[n,k]
//                                   + sum_k A2[m,k]*W2[n,k] + bias[n] )
// One wave computes one 16x16 output tile with V_WMMA_F32_16X16X4_F32.
//
// fp32 16x16x4 operand layout (ISA 7.12.2):
//   A (16x4): lane = M (mod 16); lanes 0-15 hold K=khalf+0/1 with khalf=0,
//             lanes 16-31 with khalf=2.  -> v2f per lane.
//   B (4x16): lane = N (mod 16); same khalf split.   B[k][n] = W[n][k].
//   C/D (16x16): VGPR j -> M = j + 8*(lane>=16), N = lane%16. -> v8f.
// EXEC is all-ones for every wave that reaches the WMMA loop (whole-wave
// early-out only); partial M tiles are zero-masked on load, guarded on store.
// ---------------------------------------------------------------------------
__global__ void __launch_bounds__(256)
sage_gemm_wmma(const float* __restrict__ A1, const float* __restrict__ A2,
               const float* __restrict__ W1, const float* __restrict__ W2,
               const float* __restrict__ bias, float* __restrict__ out,
               int M, int N, int tiles_n, int total_tiles)
{
    int wave = (int)((blockIdx.x * blockDim.x + threadIdx.x) >> 5);
    if (wave >= total_tiles) return;            // uniform within the wave
    int lane  = threadIdx.x & 31;
    int tm    = wave / tiles_n;
    int tn    = wave % tiles_n;

    int lmod  = lane & 15;
    int hi    = lane >> 4;                      // 0 | 1
    int khalf = hi << 1;                        // 0 | 2

    int m = tm * 16 + lmod;                     // A row this lane loads
    int n = tn * 16 + lmod;                     // B column this lane loads
    float amask = (m < M) ? 1.0f : 0.0f;
    int   mc    = (m < M) ? m : (M - 1);

    const float* a1 = A1 + (size_t)mc * FEAT + khalf;
    const float* a2 = A2 + (size_t)mc * FEAT + khalf;
    const float* w1 = W1 + (size_t)n  * FEAT + khalf;
    const float* w2 = W2 + (size_t)n  * FEAT + khalf;

    v8f acc = {};

    #pragma unroll 4
    for (int k = 0; k < FEAT; k += 4) {
        v2f a, b;
        a[0] = a1[k]     * amask;
        a[1] = a1[k + 1] * amask;
        b[0] = w1[k];
        b[1] = w1[k + 1];
        acc = __builtin_amdgcn_wmma_f32_16x16x4_f32(
            /*neg_a=*/false, a, /*neg_b=*/false, b,
            /*c_mod=*/(short)0, acc, /*reuse_a=*/false, /*reuse_b=*/false);
    }
    #pragma unroll 4
    for (int k = 0; k < FEAT; k += 4) {
        v2f a, b;
        a[0] = a2[k]     * amask;
        a[1] = a2[k + 1] * amask;
        b[0] = w2[k];
        b[1] = w2[k + 1];
        acc = __builtin_amdgcn_wmma_f32_16x16x4_f32(
            false, a, false, b, (short)0, acc, false, false);
    }

    float bn    = bias[n];
    int   mbase = tm * 16 + hi * 8;
    #pragma unroll
    for (int j = 0; j < 8; ++j) {
        int mm = mbase + j;
        if (mm < M) {
            float v = acc[j] + bn;
            out[(size_t)mm * N + n] = v > 0.0f ? v : 0.0f;
        }
    }
}

// ---------------------------------------------------------------------------
// Launcher: zero accumulators, scatter1, mean1, gemm1 -> h,
//           scatter2, mean2, gemm2 -> d_out.  All graph-capture safe.
// ---------------------------------------------------------------------------
extern "C" void kernel_launch(void* const* d_in, const int* in_sizes, int n_in,
                              void* d_out, int out_size, void* d_ws, size_t ws_size,
                              hipStream_t stream) {
    const float* x    = (const float*)d_in[0];
    const float* W1l  = (const float*)d_in[1];
    const float* b1l  = (const float*)d_in[2];
    const float* W1r  = (const float*)d_in[3];
    const float* W2l  = (const float*)d_in[4];
    const float* b2l  = (const float*)d_in[5];
    const float* W2r  = (const float*)d_in[6];
    const int*   src1 = (const int*)d_in[7];
    const int*   dst1 = (const int*)d_in[8];
    const int*   src2 = (const int*)d_in[9];
    const int*   dst2 = (const int*)d_in[10];

    const int E1     = in_sizes[7];
    const int E2     = in_sizes[9];
    const int N1FEAT = in_sizes[2];              // 128 (b1_l length)
    const int N2FEAT = in_sizes[5];              // 64  (b2_l length)
    const int SIZE2  = out_size / N2FEAT;        // 5000
    const int SIZE1  = 50000;                    // device scalar; fixed by setup

    // Workspace layout (floats): agg1 | cnt1 | h | agg2 | cnt2  (~54 MB)
    float* agg1 = (float*)d_ws;
    float* cnt1 = agg1 + (size_t)SIZE1 * FEAT;
    float* h    = cnt1 + SIZE1;
    float* agg2 = h    + (size_t)SIZE1 * FEAT;
    float* cnt2 = agg2 + (size_t)SIZE2 * FEAT;

    hipMemsetAsync(agg1, 0, ((size_t)SIZE1 * FEAT + SIZE1) * sizeof(float), stream);
    hipMemsetAsync(agg2, 0, ((size_t)SIZE2 * FEAT + SIZE2) * sizeof(float), stream);

    // ---- Layer 1 ----
    sage_scatter<<<(E1 + 7) / 8, 256, 0, stream>>>(x, src1, dst1, agg1, cnt1, E1);
    {
        int total = SIZE1 * FEAT;
        sage_mean_div<<<(total + 255) / 256, 256, 0, stream>>>(agg1, cnt1, total);
    }
    {
        int tiles_n = N1FEAT / 16;
        int tiles_m = (SIZE1 + 15) / 16;
        int total   = tiles_m * tiles_n;
        sage_gemm_wmma<<<(total + 7) / 8, 256, 0, stream>>>(
            agg1, x, W1l, W1r, b1l, h, SIZE1, N1FEAT, tiles_n, total);
    }

    // ---- Layer 2 ----
    sage_scatter<<<(E2 + 7) / 8, 256, 0, stream>>>(h, src2, dst2, agg2, cnt2, E2);
    {
        int total = SIZE2 * FEAT;
        sage_mean_div<<<(total + 255) / 256, 256, 0, stream>>>(agg2, cnt2, total);
    }
    {
        int tiles_n = N2FEAT / 16;
        int tiles_m = (SIZE2 + 15) / 16;
        int total   = tiles_m * tiles_n;
        sage_gemm_wmma<<<(total + 7) / 8, 256, 0, stream>>>(
            agg2, h, W2l, W2r, b2l, (float*)d_out, SIZE2, N2FEAT, tiles_n, total);
    }
}